// MultiHeadAttentionLayer_31129922961568
// MI455X (gfx1250) — compile-verified
//
#include <hip/hip_runtime.h>

// MI455X / gfx1250, wave32. fp32 attention layer restructured as:
//   pass1: G_b = X_b X_b^T (Gram, WMMA f32 16x16x4), s_b = X_b 1
//   pass2: scores/softmax from (G,s) analytically; collapse attn+W_v+W_out
//          into per-batch M_b (128x128), c_b (128)
//   pass3: y = M_b X_b + c_b   (WMMA f32 GEMM)
// ~400 MB HBM traffic -> ~17us floor @ 23.3 TB/s; ~18 GFLOP fp32 hidden.

typedef __attribute__((ext_vector_type(2))) float v2f;
typedef __attribute__((ext_vector_type(4))) float v4f;
typedef __attribute__((ext_vector_type(8))) float v8f;

#define C_   128
#define NH   8
#define BB   16
#define NN   16384          // tokens per batch (128*128)
#define K1_STRIDE 68        // LDS row stride (floats): 4 mod 64 -> conflict-free b64 frags
#define K4_STRIDE 80        // LDS row stride (floats): 16 mod 64 -> conflict-free b32 frags

// ---------------------------------------------------------------- pass 1: Gram
// grid: 16 batches * 16 chunks = 256 WGs; block 256 (8 waves)
// each WG: 1024 tokens in 16 stages of 64; wave w owns row-block i=w, cols j=0..7
__global__ __launch_bounds__(256) void gram_kernel(const float* __restrict__ x,
                                                   float* __restrict__ Gp,
                                                   float* __restrict__ sp) {
    __shared__ float ls[128 * K1_STRIDE];
    const int b     = blockIdx.x >> 4;
    const int chunk = blockIdx.x & 15;
    const int tid   = threadIdx.x;
    const int lane  = tid & 31;
    const int wave  = tid >> 5;
    const int lrow  = lane & 15;
    const int lhi   = lane >> 4;        // 0: K/M low half, 1: high half
    const int ldrow = tid >> 1;         // loader row (2 threads per row)
    const int ldhalf = tid & 1;

    const float* xb = x + (size_t)b * C_ * NN;
    float rowsum = 0.f;

    v8f acc[8] = {};

    for (int stage = 0; stage < 16; ++stage) {
        const int n0 = chunk * 1024 + stage * 64;
        __syncthreads();
        // stage 128 rows x 64 tokens into LDS (coalesced b128 loads)
        const float* src = xb + (size_t)ldrow * NN + n0 + ldhalf * 32;
        float*       dst = ls + ldrow * K1_STRIDE + ldhalf * 32;
#pragma unroll
        for (int j = 0; j < 8; ++j) {
            v4f v = *(const v4f*)(src + 4 * j);
            *(v4f*)(dst + 4 * j) = v;
            rowsum += v.x + v.y + v.z + v.w;
        }
        __syncthreads();
#pragma unroll 4
        for (int ks = 0; ks < 16; ++ks) {   // 4 tokens per step
            // fragment for row-block p: A and B layouts coincide for Gram
            v2f frag[8];
#pragma unroll
            for (int p = 0; p < 8; ++p)
                frag[p] = *(const v2f*)(ls + (16 * p + lrow) * K1_STRIDE + 4 * ks + 2 * lhi);
            v2f fragA = *(const v2f*)(ls + (16 * wave + lrow) * K1_STRIDE + 4 * ks + 2 * lhi);
#pragma unroll
            for (int j = 0; j < 8; ++j)
                acc[j] = __builtin_amdgcn_wmma_f32_16x16x4_f32(
                    false, fragA, false, frag[j], (short)0, acc[j], false, false);
        }
    }

    // partial row sums: combine the 2 loader threads per row via LDS
    __syncthreads();
    ls[tid] = rowsum;
    __syncthreads();
    if ((tid & 1) == 0)
        sp[((size_t)b * 16 + chunk) * 128 + ldrow] = rowsum + ls[tid + 1];

    // partial Gram: plain stores (deterministic; reduced later)
    float* gdst = Gp + ((size_t)b * 16 + chunk) * (128 * 128);
    const int grow = 16 * wave + lhi * 8;
#pragma unroll
    for (int j = 0; j < 8; ++j)
#pragma unroll
        for (int r = 0; r < 8; ++r)
            gdst[(grow + r) * 128 + 16 * j + lrow] = acc[j][r];
}

// ------------------------------------------------------- pass 1b: reduce parts
__global__ __launch_bounds__(256) void reduce_kernel(const float* __restrict__ Gp,
                                                     const float* __restrict__ sp,
                                                     float* __restrict__ G,
                                                     float* __restrict__ s) {
    int i = blockIdx.x * 256 + threadIdx.x;
    if (i < BB * 128 * 128) {
        int b = i >> 14, cc = i & 16383;
        float a = 0.f;
#pragma unroll
        for (int ch = 0; ch < 16; ++ch)
            a += Gp[(((size_t)b * 16 + ch) << 14) + cc];
        G[i] = a;
    } else if (i < BB * 128 * 128 + BB * 128) {
        int j = i - BB * 128 * 128;
        int b = j >> 7, c = j & 127;
        float a = 0.f;
#pragma unroll
        for (int ch = 0; ch < 16; ++ch)
            a += sp[(b * 16 + ch) * 128 + c];
        s[j] = a;
    }
}

// ------------------------------------------- pass 2a: scores + softmax -> attn
// grid: B*NH = 128 blocks; block 256
__global__ __launch_bounds__(256) void scores_kernel(const float* __restrict__ G,
                                                     const float* __restrict__ s,
                                                     const float* __restrict__ w_qkv,
                                                     const float* __restrict__ b_qkv,
                                                     float* __restrict__ attn) {
    __shared__ float KG[16 * 128];
    __shared__ float S[16 * 16];
    __shared__ float sk[16], sq[16];
    const int bh = blockIdx.x, b = bh >> 3, h = bh & 7;
    const int tid = threadIdx.x;
    const float* Gb = G + (size_t)b * C_ * C_;
    const float* Wq = w_qkv + (h * 16) * C_;
    const float* Wk = w_qkv + (C_ + h * 16) * C_;
    const float* sb = s + b * C_;

    // KG[e][c] = sum_c' Wk[e][c'] G[c][c']   (G symmetric)
#pragma unroll
    for (int o = 0; o < 8; ++o) {
        int idx = tid + 256 * o;
        int e = idx >> 7, c = idx & 127;
        const float* gr = Gb + c * C_;
        const float* wk = Wk + e * C_;
        float a = 0.f;
        for (int cc = 0; cc < 128; ++cc) a += wk[cc] * gr[cc];
        KG[e * 128 + c] = a;
    }
    if (tid < 32) {
        int e = tid & 15;
        const float* w = (tid < 16) ? (Wk + e * C_) : (Wq + e * C_);
        float a = 0.f;
        for (int cc = 0; cc < 128; ++cc) a += w[cc] * sb[cc];
        if (tid < 16) sk[e] = a; else sq[e] = a;
    }
    __syncthreads();
    {
        int d = tid >> 4, e = tid & 15;
        const float* wq = Wq + d * C_;
        float a = 0.f;
        for (int cc = 0; cc < 128; ++cc) a += wq[cc] * KG[e * 128 + cc];
        float bq = b_qkv[h * 16 + d];
        float bk = b_qkv[C_ + h * 16 + e];
        a += sq[d] * bk + bq * sk[e] + (float)NN * bq * bk;
        S[d * 16 + e] = a * 0.08838834764831845f;   // 1/sqrt(128)
    }
    __syncthreads();
    if (tid < 16) {
        int d = tid;
        float m = -1e30f;
#pragma unroll
        for (int e = 0; e < 16; ++e) m = fmaxf(m, S[d * 16 + e]);
        float ex[16], sum = 0.f;
#pragma unroll
        for (int e = 0; e < 16; ++e) { ex[e] = __expf(S[d * 16 + e] - m); sum += ex[e]; }
        float inv = 1.f / sum;
#pragma unroll
        for (int e = 0; e < 16; ++e)
            attn[((size_t)bh * 16 + d) * 16 + e] = ex[e] * inv;
    }
}

// --------------------------------- pass 2b: M_b = W_out A_b W_v ; c_b
// grid: B = 16 blocks; block 256
__global__ __launch_bounds__(256) void compose_kernel(const float* __restrict__ attn,
                                                      const float* __restrict__ w_qkv,
                                                      const float* __restrict__ b_qkv,
                                                      const float* __restrict__ w_out,
                                                      const float* __restrict__ b_out,
                                                      float* __restrict__ M,
                                                      float* __restrict__ cvec) {
    __shared__ float U[128 * 129];   // stride 129 -> conflict-free column reads
    __shared__ float ub[128];
    const int b = blockIdx.x, tid = threadIdx.x;
    const float* Wv = w_qkv + 2 * C_ * C_;
    const float* bv = b_qkv + 2 * C_;
    const float* A  = attn + (size_t)b * NH * 256;

    // U[c][c'] = sum_e attn[h, dd, e] * Wv[h*16+e][c'],  c = h*16+dd
#pragma unroll
    for (int o = 0; o < 64; ++o) {
        int idx = tid + 256 * o;
        int c = idx >> 7, cp = idx & 127;
        int h = c >> 4;
        const float* a = A + c * 16;   // (h*16+dd)*16
        float acc = 0.f;
#pragma unroll
        for (int e = 0; e < 16; ++e) acc += a[e] * Wv[(h * 16 + e) * C_ + cp];
        U[c * 129 + cp] = acc;
    }
    if (tid < 128) {
        int c = tid, h = c >> 4;
        const float* a = A + c * 16;
        float acc = 0.f;
#pragma unroll
        for (int e = 0; e < 16; ++e) acc += a[e] * bv[h * 16 + e];
        ub[c] = acc;
    }
    __syncthreads();
#pragma unroll
    for (int o = 0; o < 64; ++o) {
        int idx = tid + 256 * o;
        int oc = idx >> 7, cp = idx & 127;
        const float* wo = w_out + oc * C_;
        float acc = 0.f;
        for (int c = 0; c < 128; ++c) acc += wo[c] * U[c * 129 + cp];
        M[((size_t)b * C_ + oc) * C_ + cp] = acc;
    }
    if (tid < 128) {
        int oc = tid;
        const float* wo = w_out + oc * C_;
        float acc = 0.f;
        for (int c = 0; c < 128; ++c) acc += wo[c] * ub[c];
        cvec[b * C_ + oc] = acc + b_out[oc];
    }
}

// ---------------------------------------------------- pass 3: y = M_b X + c_b
// grid: 16 batches * 32 chunks = 512 WGs; block 256 (8 waves)
// wave w owns output row-block 16w; A-frags of M preloaded in 64 VGPRs
__global__ __launch_bounds__(256) void apply_kernel(const float* __restrict__ x,
                                                    const float* __restrict__ M,
                                                    const float* __restrict__ cvec,
                                                    float* __restrict__ y) {
    __shared__ float ls[128 * K4_STRIDE];
    const int b     = blockIdx.x >> 5;
    const int chunk = blockIdx.x & 31;     // 512 tokens each
    const int tid   = threadIdx.x;
    const int lane  = tid & 31;
    const int wave  = tid >> 5;
    const int lrow  = lane & 15;
    const int lhi   = lane >> 4;
    const int ldrow = tid >> 1;
    const int ldhalf = tid & 1;

    const float* xb = x + (size_t)b * C_ * NN;
    const float* Mb = M + (size_t)b * C_ * C_;
    float*       yb = y + (size_t)b * C_ * NN;

    v2f fa[32];
#pragma unroll
    for (int ks = 0; ks < 32; ++ks)
        fa[ks] = *(const v2f*)(Mb + (16 * wave + lrow) * C_ + 4 * ks + 2 * lhi);
    float cc[8];
#pragma unroll
    for (int r = 0; r < 8; ++r) cc[r] = cvec[b * C_ + 16 * wave + lhi * 8 + r];

    for (int stage = 0; stage < 8; ++stage) {
        const int n0 = chunk * 512 + stage * 64;
        __syncthreads();
        const float* src = xb + (size_t)ldrow * NN + n0 + ldhalf * 32;
        float*       dst = ls + ldrow * K4_STRIDE + ldhalf * 32;
#pragma unroll
        for (int j = 0; j < 8; ++j)
            *(v4f*)(dst + 4 * j) = *(const v4f*)(src + 4 * j);
        __syncthreads();
#pragma unroll
        for (int tt = 0; tt < 4; ++tt) {    // 4 token-tiles of 16
            v8f acc = {};
#pragma unroll
            for (int ks = 0; ks < 32; ++ks) {   // K over 128 channels
                v2f fb;
                fb.x = ls[(4 * ks + 2 * lhi)     * K4_STRIDE + 16 * tt + lrow];
                fb.y = ls[(4 * ks + 2 * lhi + 1) * K4_STRIDE + 16 * tt + lrow];
                acc = __builtin_amdgcn_wmma_f32_16x16x4_f32(
                    false, fa[ks], false, fb, (short)0, acc, false, false);
            }
#pragma unroll
            for (int r = 0; r < 8; ++r)
                yb[(size_t)(16 * wave + lhi * 8 + r) * NN + n0 + 16 * tt + lrow]
                    = acc[r] + cc[r];
        }
    }
}

// ---------------------------------------------------------------------- launch
extern "C" void kernel_launch(void* const* d_in, const int* in_sizes, int n_in,
                              void* d_out, int out_size, void* d_ws, size_t ws_size,
                              hipStream_t stream) {
    const float* x      = (const float*)d_in[0];
    const float* w_qkv  = (const float*)d_in[1];
    const float* b_qkv  = (const float*)d_in[2];
    const float* w_out  = (const float*)d_in[3];
    const float* b_out  = (const float*)d_in[4];
    float* y  = (float*)d_out;
    float* ws = (float*)d_ws;

    // workspace layout (floats): ~19.2 MB total
    float* Gp   = ws;                          // 16*16*16384 partial Grams
    float* sp   = Gp + (size_t)BB * 16 * 16384;   // 16*16*128 partial row sums
    float* G    = sp + BB * 16 * 128;          // 16*128*128
    float* s    = G + BB * 128 * 128;          // 16*128
    float* attn = s + BB * 128;                // 16*8*16*16
    float* M    = attn + BB * NH * 256;        // 16*128*128
    float* cvec = M + BB * 128 * 128;          // 16*128

    gram_kernel<<<256, 256, 0, stream>>>(x, Gp, sp);
    int nred = BB * 128 * 128 + BB * 128;
    reduce_kernel<<<(nred + 255) / 256, 256, 0, stream>>>(Gp, sp, G, s);
    scores_kernel<<<BB * NH, 256, 0, stream>>>(G, s, w_qkv, b_qkv, attn);
    compose_kernel<<<BB, 256, 0, stream>>>(attn, w_qkv, b_qkv, w_out, b_out, M, cvec);
    apply_kernel<<<BB * 32, 256, 0, stream>>>(x, M, cvec, y);
}